// PhysicalBasis_1932735283618
// MI455X (gfx1250) — compile-verified
//
#include <hip/hip_runtime.h>

typedef __attribute__((ext_vector_type(2))) float v2f;
typedef __attribute__((ext_vector_type(8))) float v8f;

#define N_TERMS 200
#define TILE_N 16
#define WAVES_PER_BLOCK 8
#define BLOCK_THREADS 256
#define PI_OVER_PERIOD 0.31415926535897932f

// out[j] = sum_m coeff[m] * basis(m, x[j])
//   l==0: basis = cos((m+0.5)*theta), l>0: basis = sin((m+1)*theta), theta = (pi/10)*x
// Chebyshev stride-4 recurrence: v(m+4) = 2*cos(4*theta)*v(m) - v(m-4)
// Contraction done with V_WMMA_F32_16X16X4_F32, 16 x-values per wave.
// Single sincosf(theta/2) per tile; all other trig values synthesized via
// angle addition + the same Chebyshev recurrence (pure FMAs).
__global__ __launch_bounds__(BLOCK_THREADS) void physbasis_wmma_kernel(
    const int* __restrict__ d_n, const int* __restrict__ d_l,
    const float* __restrict__ d_x, const float* __restrict__ d_eig,
    float* __restrict__ d_out, int n_x, int num_tiles)
{
    __shared__ float s_coeff[N_TERMS];

    const int tid  = threadIdx.x;
    const int nIdx = d_n[0];
    const int lIdx = d_l[0];

    // coeff[m] = eigenvectors[l][m][n] -> eig[l*200*200 + m*200 + n]; stage in LDS
    const float* eigL = d_eig + (size_t)lIdx * N_TERMS * N_TERMS + nIdx;
    for (int m = tid; m < N_TERMS; m += BLOCK_THREADS)
        s_coeff[m] = eigL[(size_t)m * N_TERMS];
    __syncthreads();

    const int wave = tid >> 5;
    const int lane = tid & 31;
    const int tile = blockIdx.x * WAVES_PER_BLOCK + wave;
    if (tile >= num_tiles) return;   // wave-uniform: EXEC stays all-1s for WMMA

    const int col  = lane & 15;      // N index (column of B / D)
    const int kgrp = lane >> 4;      // lane half selects K pair {0,1} vs {2,3}

    int xi = tile * TILE_N + col;
    if (xi >= n_x) xi = n_x - 1;     // clamp load (store is guarded separately)
    const float theta = PI_OVER_PERIOD * d_x[xi];

    // --- single transcendental: sincos of theta/2 ---
    float sh, ch;                    // sin(theta/2), cos(theta/2)
    sincosf(0.5f * theta, &sh, &ch);
    const float c   = fmaf(-2.0f * sh, sh, 1.0f);   // cos(theta)
    const float s   = 2.0f * sh * ch;               // sin(theta)
    const float tc  = 2.0f * c;                     // 2*cos(theta)
    const float c2  = fmaf(tc, c, -1.0f);           // cos(2*theta)
    const float k2  = 2.0f * fmaf(2.0f * c2, c2, -1.0f);   // 2*cos(4*theta)

    // Two interleaved stride-4 recurrences per lane:
    //   lane half kgrp=0 covers m offsets {0,1}, kgrp=1 covers {2,3}
    float cur0, cur1, prev0, prev1;
    if (lIdx == 0) {
        // v_k = cos((k+0.5)*theta); v_{-1} = v_0 = ch; cos even => v(-k-1) = v_k
        const float v0 = ch;
        const float v1 = fmaf(tc, v0, -v0);
        const float v2 = fmaf(tc, v1, -v0);
        const float v3 = fmaf(tc, v2, -v1);
        cur0  = kgrp ? v2 : v0;   // m = base
        cur1  = kgrp ? v3 : v1;   // m = base+1
        prev0 = kgrp ? v1 : v3;   // m = base-4
        prev1 = kgrp ? v0 : v2;   // m = base-3
    } else {
        // s_k = sin(k*theta); basis(m) = s_{m+1}; sin odd => s_{-k} = -s_k
        const float s1 = s;
        const float s2 = tc * s;
        const float s3 = fmaf(tc, s2, -s1);
        const float s4 = fmaf(tc, s3, -s2);
        cur0  = kgrp ?  s3 :  s1;
        cur1  = kgrp ?  s4 :  s2;
        prev0 = kgrp ? -s1 : -s3;
        prev1 = kgrp ? 0.0f : -s2;
    }

    const int base = kgrp * 2;       // K offset within current 4-wide chunk
    v8f acc = {};
    for (int t = 0; t < N_TERMS / 4; ++t) {
        const int m = 4 * t + base;
        // A-matrix (16x4): every M-row identical = coeff chunk
        v2f a; a[0] = s_coeff[m]; a[1] = s_coeff[m + 1];
        // B-matrix (4x16): basis values for column `col`, rows K=base, base+1
        v2f b; b[0] = cur0; b[1] = cur1;
        acc = __builtin_amdgcn_wmma_f32_16x16x4_f32(
            false, a, false, b, (short)0, acc, false, false);
        // advance both recurrences by 4 in m
        float nx0 = fmaf(k2, cur0, -prev0);
        float nx1 = fmaf(k2, cur1, -prev1);
        prev0 = cur0; cur0 = nx0;
        prev1 = cur1; cur1 = nx1;
    }

    // D layout: VGPR0, lanes 0-15 hold row M=0 (all rows equal) for N=lane
    const int oi = tile * TILE_N + col;
    if (kgrp == 0 && oi < n_x)
        d_out[oi] = acc[0];
}

extern "C" void kernel_launch(void* const* d_in, const int* in_sizes, int n_in,
                              void* d_out, int out_size, void* d_ws, size_t ws_size,
                              hipStream_t stream) {
    const int*   d_n   = (const int*)d_in[0];
    const int*   d_l   = (const int*)d_in[1];
    const float* d_x   = (const float*)d_in[2];
    const float* d_eig = (const float*)d_in[3];
    float*       out   = (float*)d_out;

    const int n_x       = in_sizes[2];
    const int num_tiles = (n_x + TILE_N - 1) / TILE_N;
    const int blocks    = (num_tiles + WAVES_PER_BLOCK - 1) / WAVES_PER_BLOCK;

    physbasis_wmma_kernel<<<blocks, BLOCK_THREADS, 0, stream>>>(
        d_n, d_l, d_x, d_eig, out, n_x, num_tiles);
}